// RegionFeedback_74088185856151
// MI455X (gfx1250) — compile-verified
//
#include <hip/hip_runtime.h>
#include <math.h>

// Problem constants from the reference: B=4, T=4096, D=1024, A=64
#define BB 4
#define TT 4096
#define DD 1024
#define AA 64

typedef __attribute__((ext_vector_type(2))) float v2f;
typedef __attribute__((ext_vector_type(8))) float v8f;

// ---------------------------------------------------------------------------
// K0: segment starts. assign is sorted, so starts[a] = #(assign[t] < a),
// starts[A] = T. One block, 128 threads, assign staged in LDS.
// ---------------------------------------------------------------------------
__global__ __launch_bounds__(128) void starts_kernel(const int* __restrict__ assign,
                                                     int* __restrict__ starts) {
    __shared__ int sh[TT];
    for (int i = threadIdx.x; i < TT; i += 128) sh[i] = assign[i];
    __syncthreads();
    int a = threadIdx.x;
    if (a <= AA) {
        int cnt = 0;
        for (int t = 0; t < TT; ++t) cnt += (sh[t] < a) ? 1 : 0;
        starts[a] = cnt;
    }
}

// ---------------------------------------------------------------------------
// K1: anchorsum[b,a,:] = sum over x rows in [starts[a], starts[a+1]).
// Grid (A, B), 256 threads, each thread owns 4 strided d-columns.
// Coalesced row reads; no atomics (sorted assign -> contiguous segments).
// ---------------------------------------------------------------------------
__global__ __launch_bounds__(256) void anchor_sum(const float* __restrict__ x,
                                                  const int* __restrict__ starts,
                                                  float* __restrict__ anchorsum) {
    const int a = blockIdx.x;
    const int b = blockIdx.y;
    const int s = starts[a];
    const int e = starts[a + 1];
    float acc0 = 0.f, acc1 = 0.f, acc2 = 0.f, acc3 = 0.f;
    const int tid = threadIdx.x;
    for (int t = s; t < e; ++t) {
        const float* row = x + ((size_t)b * TT + t) * DD;
        acc0 += row[tid];
        acc1 += row[tid + 256];
        acc2 += row[tid + 512];
        acc3 += row[tid + 768];
    }
    float* orow = anchorsum + ((size_t)b * AA + a) * DD;
    orow[tid]       = acc0;
    orow[tid + 256] = acc1;
    orow[tid + 512] = acc2;
    orow[tid + 768] = acc3;
}

// ---------------------------------------------------------------------------
// K2: normalize by counts, then `hops` rounds of agg = W_graph @ agg.
// Block handles one b and a 16-wide d slice; all A=64 anchors live in LDS so
// the hop loop needs only block-level syncs. W_graph (16 KB) cached in LDS.
// ---------------------------------------------------------------------------
__global__ __launch_bounds__(1024) void graph_kernel(const float* __restrict__ anchorsum,
                                                     const float* __restrict__ counts,
                                                     const float* __restrict__ Wg,
                                                     const int* __restrict__ hops_p,
                                                     float* __restrict__ agg) {
    __shared__ float Ws[AA * AA];     // 16 KB
    __shared__ float cur[AA * 16];    // 4 KB
    __shared__ float nxt[AA * 16];    // 4 KB
    const int tid = threadIdx.x;              // 0..1023
    const int b  = blockIdx.y;                // 0..3
    const int d0 = blockIdx.x * 16;           // 64 slices
    for (int i = tid; i < AA * AA; i += 1024) Ws[i] = Wg[i];
    const int d = tid & 15;
    const int c = tid >> 4;                   // anchor index (also output row a)
    cur[c * 16 + d] = anchorsum[((size_t)b * AA + c) * DD + d0 + d] / counts[c];
    __syncthreads();
    int hops = hops_p[0];
    if (hops < 1) hops = 1;
    for (int h = 0; h < hops; ++h) {
        float acc = 0.f;
        #pragma unroll 8
        for (int cc = 0; cc < AA; ++cc)
            acc += Ws[c * AA + cc] * cur[cc * 16 + d];
        nxt[c * 16 + d] = acc;
        __syncthreads();
        cur[c * 16 + d] = nxt[c * 16 + d];
        __syncthreads();
    }
    agg[((size_t)b * AA + c) * DD + d0 + d] = cur[c * 16 + d];
}

// ---------------------------------------------------------------------------
// K3: proj[r, n] = sum_k agg[r, k] * W_proj[n, k]   (M=256, N=1024, K=1024)
// One wave per 16x16 output tile using V_WMMA_F32_16X16X4_F32.
// f32 A-matrix 16x4 layout (ISA 7.12.2): lanes 0-15 hold (M=lane, K=0..1),
// lanes 16-31 hold (M=lane-16, K=2..3). B 4x16 mirrors on K halves.
// C/D: VGPR i -> (M=i, N=lane) for lanes 0-15, (M=i+8, N=lane-16) for 16-31.
// ---------------------------------------------------------------------------
__global__ __launch_bounds__(32) void proj_wmma(const float* __restrict__ agg,
                                                const float* __restrict__ Wp,
                                                float* __restrict__ proj) {
    const int tileId = blockIdx.x;     // 1024 tiles: 16 along M x 64 along N
    const int tm = tileId & 15;
    const int tn = tileId >> 4;
    const int lane = threadIdx.x;
    const int half = lane >> 4;        // K sub-half: 0 -> K=0..1, 1 -> K=2..3
    const int l = lane & 15;
    const float* aPtr = agg + ((size_t)(tm * 16 + l)) * DD + half * 2;
    const float* bPtr = Wp  + ((size_t)(tn * 16 + l)) * DD + half * 2;
    v8f acc = {};
    #pragma unroll 4
    for (int k = 0; k < DD; k += 4) {
        v2f av = *(const v2f*)(aPtr + k);
        v2f bv = *(const v2f*)(bPtr + k);
        acc = __builtin_amdgcn_wmma_f32_16x16x4_f32(
            /*neg_a=*/false, av, /*neg_b=*/false, bv,
            /*c_mod=*/(short)0, acc, /*reuse_a=*/false, /*reuse_b=*/false);
    }
    float* outBase = proj + (size_t)(tm * 16) * DD + tn * 16 + l;
    #pragma unroll
    for (int i = 0; i < 8; ++i) {
        const int m = i + half * 8;
        outBase[(size_t)m * DD] = acc[i];
    }
}

// ---------------------------------------------------------------------------
// K4: out[b,t,:] = x[b,t,:] + proj[b, assign[t], :] * tanh(gate)
// float4-vectorized; this is the bandwidth-dominant pass (64MB in + 64MB out,
// plus proj gather which stays resident in L2).
// ---------------------------------------------------------------------------
__global__ __launch_bounds__(256) void out_kernel(const float4* __restrict__ x,
                                                  const float* __restrict__ proj,
                                                  const int* __restrict__ assign,
                                                  const float* __restrict__ gate,
                                                  float4* __restrict__ out) {
    const int idx = blockIdx.x * 256 + threadIdx.x;   // 0 .. B*T*D/4-1
    const int d4 = idx & (DD / 4 - 1);                // 0..255
    const int bt = idx >> 8;                          // 0..16383
    const int t  = bt & (TT - 1);
    const int b  = bt >> 12;
    const int a  = assign[t];
    const float g = tanhf(gate[0]);
    const float4 xv = x[idx];
    const float4 pv = ((const float4*)proj)[((size_t)b * AA + a) * (DD / 4) + d4];
    float4 r;
    r.x = xv.x + pv.x * g;
    r.y = xv.y + pv.y * g;
    r.z = xv.z + pv.z * g;
    r.w = xv.w + pv.w * g;
    out[idx] = r;
}

// ---------------------------------------------------------------------------
extern "C" void kernel_launch(void* const* d_in, const int* in_sizes, int n_in,
                              void* d_out, int out_size, void* d_ws, size_t ws_size,
                              hipStream_t stream) {
    const float* x      = (const float*)d_in[0];
    const float* Wp     = (const float*)d_in[1];
    const float* gate   = (const float*)d_in[2];
    const int*   assign = (const int*)d_in[3];
    const float* Wg     = (const float*)d_in[4];
    const float* counts = (const float*)d_in[5];
    const int*   hops   = (const int*)d_in[6];
    float* out = (float*)d_out;

    char* ws = (char*)d_ws;
    float* anchorsum = (float*)(ws);                    // B*A*D f32 = 1 MB
    float* agg       = (float*)(ws + (1u << 20));       // 1 MB
    float* proj      = (float*)(ws + (2u << 20));       // 1 MB
    int*   starts    = (int*)  (ws + (3u << 20));       // 65 ints

    starts_kernel<<<dim3(1), dim3(128), 0, stream>>>(assign, starts);
    anchor_sum<<<dim3(AA, BB), dim3(256), 0, stream>>>(x, starts, anchorsum);
    graph_kernel<<<dim3(DD / 16, BB), dim3(1024), 0, stream>>>(anchorsum, counts, Wg, hops, agg);
    proj_wmma<<<dim3((256 / 16) * (DD / 16)), dim3(32), 0, stream>>>(agg, Wp, proj);
    out_kernel<<<dim3((BB * TT * DD / 4) / 256), dim3(256), 0, stream>>>(
        (const float4*)x, proj, assign, gate, (float4*)out);
}